// SimpleConv_62079457296944
// MI455X (gfx1250) — compile-verified
//
#include <hip/hip_runtime.h>

// GCN 2-layer forward for MI455X (gfx1250, wave32).
// Memory-bound: ~200MB HBM traffic -> ~9us at 23.3 TB/s; feature tables
// (6.4MB) are L2-resident so gather/atomic-scatter hit the 192MB L2.
// GEMMs use V_WMMA_F32_16X16X4_F32 (fp32-exact vs. reference).

typedef __attribute__((ext_vector_type(2))) float v2f;
typedef __attribute__((ext_vector_type(8))) float v8f;

#define F_IN 128
#define DIM  16

// ---------------- degree / normalization ----------------

__global__ void k_init_deg(int* __restrict__ deg, int n) {
    int i = blockIdx.x * blockDim.x + threadIdx.x;
    if (i < n) deg[i] = 1;  // self-loop
}

__global__ void k_count_deg(const int* __restrict__ col, int* __restrict__ deg, int e) {
    int i = blockIdx.x * blockDim.x + threadIdx.x;
    if (i < e) atomicAdd(&deg[col[i]], 1);
}

__global__ void k_dinv(const int* __restrict__ deg, float* __restrict__ dinv, int n) {
    int i = blockIdx.x * blockDim.x + threadIdx.x;
    if (i < n) {
        float d = (float)deg[i];
        dinv[i] = (d > 0.0f) ? rsqrtf(d) : 0.0f;
    }
}

// ---------------- WMMA GEMM: H[n,16] = X[n,K] @ W[K,16] ----------------
// One wave computes a 16x16 output tile with v_wmma_f32_16x16x4_f32.
// A frag (16x4 f32): lane L -> row L&15, Kpair (L>>4)*2 + {0,1}  (2 VGPRs)
// B frag (4x16 f32): lane L -> col L&15, Kpair (L>>4)*2 + {0,1}  (2 VGPRs)
// C/D   (16x16 f32): vgpr j -> row j + (L>>4)*8, col L&15        (8 VGPRs)

template <int K, int PITCH>
__global__ void k_gemm16(const float* __restrict__ X, const float* __restrict__ W,
                         float* __restrict__ H, int n) {
    int wave = (int)((blockIdx.x * blockDim.x + threadIdx.x) >> 5);
    int lane = threadIdx.x & 31;
    int m0 = wave * 16;
    if (m0 >= n) return;                       // wave-uniform: EXEC all-ones below

    int mrow = m0 + (lane & 15);
    if (mrow >= n) mrow = n - 1;               // clamp; clamped rows never stored
    const int khalf = (lane >> 4) * 2;
    const int ncol  = lane & 15;

    v8f acc = {0.f, 0.f, 0.f, 0.f, 0.f, 0.f, 0.f, 0.f};

#pragma unroll
    for (int k = 0; k < K; k += 4) {
        v2f a, b;
        const float* xp = X + (size_t)mrow * PITCH + (k + khalf);
        a.x = xp[0];
        a.y = xp[1];
        const float* wp = W + (size_t)(k + khalf) * DIM + ncol;
        b.x = wp[0];
        b.y = wp[DIM];
        acc = __builtin_amdgcn_wmma_f32_16x16x4_f32(
            /*neg_a=*/false, a, /*neg_b=*/false, b,
            /*c_mod=*/(short)0, acc, /*reuse_a=*/false, /*reuse_b=*/false);
    }

    int rbase = m0 + (lane >> 4) * 8;
#pragma unroll
    for (int j = 0; j < 8; ++j) {
        int r = rbase + j;
        if (r < n) H[(size_t)r * DIM + ncol] = acc[j];
    }
}

// ---------------- aggregation ----------------

// self-loop contribution: acc[i,f] = dinv[i]^2 * h[i,f]
__global__ void k_self_init(const float* __restrict__ h, const float* __restrict__ dinv,
                            float* __restrict__ acc, int n) {
    size_t t = (size_t)blockIdx.x * blockDim.x + threadIdx.x;
    if (t < (size_t)n * DIM) {
        int i = (int)(t >> 4);
        float di = dinv[i];
        acc[t] = di * di * h[t];
    }
}

// edge messages: acc[col,f] += h[row,f] * dinv[row]*dinv[col]
// 16 lanes per edge -> contiguous 64B feature rows, coalesced; atomics L2-resident.
__global__ void k_scatter(const int* __restrict__ row, const int* __restrict__ col,
                          const float* __restrict__ h, const float* __restrict__ dinv,
                          float* __restrict__ acc, int e) {
    long long t = (long long)blockIdx.x * blockDim.x + threadIdx.x;
    long long idx = t >> 4;
    int f = (int)(t & 15);
    if (idx >= e) return;
    int r = row[idx];
    int c = col[idx];
    float w = dinv[r] * dinv[c];
    atomicAdd(&acc[(size_t)c * DIM + f], h[(size_t)r * DIM + f] * w);
}

__global__ void k_bias_relu(const float* __restrict__ acc, const float* __restrict__ bias,
                            float* __restrict__ out, int n) {
    size_t t = (size_t)blockIdx.x * blockDim.x + threadIdx.x;
    if (t < (size_t)n * DIM) {
        float v = acc[t] + bias[t & 15];
        out[t] = v > 0.0f ? v : 0.0f;
    }
}

// ---------------- launch ----------------

extern "C" void kernel_launch(void* const* d_in, const int* in_sizes, int n_in,
                              void* d_out, int out_size, void* d_ws, size_t ws_size,
                              hipStream_t stream) {
    const float* x  = (const float*)d_in[0];
    const int*   ei = (const int*)d_in[1];
    const float* W1 = (const float*)d_in[2];
    const float* b1 = (const float*)d_in[3];
    const float* W2 = (const float*)d_in[4];
    const float* b2 = (const float*)d_in[5];

    const int n = in_sizes[0] / F_IN;   // 100000
    const int e = in_sizes[1] / 2;      // 3200000
    const int* row = ei;
    const int* col = ei + e;

    // workspace carve-up (floats): deg(int,N) | dinv(N) | h(16N) | tmp(16N) | act(16N)
    float* ws   = (float*)d_ws;
    int*   deg  = (int*)ws;
    float* dinv = ws + n;
    float* h    = dinv + n;
    float* tmp  = h + (size_t)n * DIM;
    float* act  = tmp + (size_t)n * DIM;

    const int B = 256;
    const int gN   = (n + B - 1) / B;
    const int gE   = (e + B - 1) / B;
    const int gNF  = (int)(((size_t)n * DIM + B - 1) / B);
    const int gEF  = (int)(((long long)e * DIM + B - 1) / B);
    const int gGem = (((n + 15) / 16) + 7) / 8;   // 8 waves/block, 16 rows/wave

    // normalization (shared by both layers)
    k_init_deg<<<gN, B, 0, stream>>>(deg, n);
    k_count_deg<<<gE, B, 0, stream>>>(col, deg, e);
    k_dinv<<<gN, B, 0, stream>>>(deg, dinv, n);

    // layer 1: h = x @ W1 ; aggregate ; +b1 ; relu -> act
    k_gemm16<F_IN, F_IN><<<gGem, B, 0, stream>>>(x, W1, h, n);
    k_self_init<<<gNF, B, 0, stream>>>(h, dinv, tmp, n);
    k_scatter<<<gEF, B, 0, stream>>>(row, col, h, dinv, tmp, e);
    k_bias_relu<<<gNF, B, 0, stream>>>(tmp, b1, act, n);

    // layer 2: h = act @ W2 ; aggregate ; +b2 ; relu -> d_out
    k_gemm16<DIM, DIM><<<gGem, B, 0, stream>>>(act, W2, h, n);
    k_self_init<<<gNF, B, 0, stream>>>(h, dinv, tmp, n);
    k_scatter<<<gEF, B, 0, stream>>>(row, col, h, dinv, tmp, e);
    k_bias_relu<<<gNF, B, 0, stream>>>(tmp, b2, (float*)d_out, n);
}